// LinkPredictor_13984413515968
// MI455X (gfx1250) — compile-verified
//
#include <hip/hip_runtime.h>
#include <math.h>

// ---------------------------------------------------------------------------
// GAT link predictor for gfx1250 (MI455X, wave32, WMMA).
// Matrix work uses V_WMMA_F32_16X16X4_F32 (full fp32 precision, matches the
// f32 reference). Segment softmax uses HW L2 atomics (agent scope, relaxed).
// The edge-MLP kernel uses persistent waves with w1 fragments register-
// resident across tiles.
// ---------------------------------------------------------------------------

typedef __attribute__((ext_vector_type(2))) float v2f;
typedef __attribute__((ext_vector_type(8))) float v8f;

#define NEG_SLOPE 0.2f
#define TEMPERATURE 0.7f

__device__ __forceinline__ unsigned fenc(float f) {
  unsigned u = __float_as_uint(f);
  return (u & 0x80000000u) ? ~u : (u | 0x80000000u);
}
__device__ __forceinline__ float fdec(unsigned u) {
  return __uint_as_float((u & 0x80000000u) ? (u & 0x7FFFFFFFu) : ~u);
}
__device__ __forceinline__ float lrelu(float x) { return x > 0.0f ? x : NEG_SLOPE * x; }

__device__ __forceinline__ v8f wmma_f32(v2f a, v2f b, v8f c) {
  // V_WMMA_F32_16X16X4_F32 : D = A(16x4) * B(4x16) + C(16x16), all fp32.
  return __builtin_amdgcn_wmma_f32_16x16x4_f32(false, a, false, b, (short)0, c,
                                               false, false);
}

// ---------------------------------------------------------------------------
// K0: re-initialize accumulators (harness poisons ws once, never re-poisons).
// ---------------------------------------------------------------------------
__global__ void k_init(float* __restrict__ acc, float* __restrict__ esum,
                       unsigned* __restrict__ emax, int N_) {
  long i = (long)blockIdx.x * blockDim.x + threadIdx.x;
  if (i < (long)N_ * 64) acc[i] = 0.0f;
  if (i < (long)N_ * 2) { esum[i] = 0.0f; emax[i] = 0u; }  // 0 < fenc(-inf)
}

// ---------------------------------------------------------------------------
// K1: xp = x @ W   ([N,16] @ [16,64]) via f32 WMMA. One 16-node tile per wave.
// A layout (16x4 f32): lanes 0-15 -> {K0,K1}, lanes 16-31 -> {K2,K3}, M=lane&15
// B layout (4x16 f32): lanes 0-15 -> rows {K0,K1}, lanes 16-31 -> {K2,K3}
// C/D (16x16 f32): vgpr r: M=r (lanes 0-15) / M=r+8 (lanes 16-31), N=lane&15
// ---------------------------------------------------------------------------
__global__ void k_xform(const float* __restrict__ x, const float* __restrict__ W,
                        float* __restrict__ xp, int N_) {
  long wave = ((long)blockIdx.x * blockDim.x + threadIdx.x) >> 5;
  int lane = threadIdx.x & 31;
  int half = lane >> 4;
  int l16 = lane & 15;
  long base = wave * 16;
  if (base >= N_) return;  // wave-uniform exit, no barriers in this kernel

  long row = base + l16;
  if (row >= N_) row = N_ - 1;  // clamp loads; stores guarded below

  v2f a[4];
#pragma unroll
  for (int t = 0; t < 4; ++t) {
    int k = 4 * t + 2 * half;
    a[t].x = x[row * 16 + k];
    a[t].y = x[row * 16 + k + 1];
  }

  v8f c0 = {}, c1 = {}, c2 = {}, c3 = {};
#pragma unroll
  for (int t = 0; t < 4; ++t) {
    int k = 4 * t + 2 * half;
    v2f b;
    b.x = W[(long)k * 64 + 0 * 16 + l16];
    b.y = W[(long)(k + 1) * 64 + 0 * 16 + l16];
    c0 = wmma_f32(a[t], b, c0);
    b.x = W[(long)k * 64 + 1 * 16 + l16];
    b.y = W[(long)(k + 1) * 64 + 1 * 16 + l16];
    c1 = wmma_f32(a[t], b, c1);
    b.x = W[(long)k * 64 + 2 * 16 + l16];
    b.y = W[(long)(k + 1) * 64 + 2 * 16 + l16];
    c2 = wmma_f32(a[t], b, c2);
    b.x = W[(long)k * 64 + 3 * 16 + l16];
    b.y = W[(long)(k + 1) * 64 + 3 * 16 + l16];
    c3 = wmma_f32(a[t], b, c3);
  }

#pragma unroll
  for (int r = 0; r < 8; ++r) {
    long m = base + r + 8 * half;
    if (m < N_) {
      xp[m * 64 + 0 * 16 + l16] = c0[r];
      xp[m * 64 + 1 * 16 + l16] = c1[r];
      xp[m * 64 + 2 * 16 + l16] = c2[r];
      xp[m * 64 + 3 * 16 + l16] = c3[r];
    }
  }
}

// ---------------------------------------------------------------------------
// K2: per-(node,head) attention coefficients a_src, a_dst.
// ---------------------------------------------------------------------------
__global__ void k_att_coef(const float* __restrict__ xp,
                           const float* __restrict__ att_src,
                           const float* __restrict__ att_dst,
                           float* __restrict__ asrc, float* __restrict__ adst,
                           int N_) {
  long i = (long)blockIdx.x * blockDim.x + threadIdx.x;
  if (i >= (long)N_ * 2) return;
  int h = (int)(i & 1);
  long n = i >> 1;
  const float* row = xp + n * 64 + h * 32;
  float s = 0.0f, t = 0.0f;
#pragma unroll
  for (int c = 0; c < 32; ++c) {
    float v = row[c];
    s += v * att_src[h * 32 + c];
    t += v * att_dst[h * 32 + c];
  }
  asrc[i] = s;
  adst[i] = t;
}

// ---------------------------------------------------------------------------
// K3: segment max of edge logits over destination (incl. self loops).
// ---------------------------------------------------------------------------
__global__ void k_att_max(const long long* __restrict__ ei,
                          const float* __restrict__ asrc,
                          const float* __restrict__ adst,
                          unsigned* __restrict__ emax, long E_, int N_) {
  long i = (long)blockIdx.x * blockDim.x + threadIdx.x;
  long M = E_ + (long)N_;
  if (i >= M) return;
  int s, d;
  if (i < E_) { s = (int)ei[i]; d = (int)ei[E_ + i]; }
  else        { s = d = (int)(i - E_); }
#pragma unroll
  for (int h = 0; h < 2; ++h) {
    float e = lrelu(asrc[(long)s * 2 + h] + adst[(long)d * 2 + h]);
    __hip_atomic_fetch_max(&emax[(long)d * 2 + h], fenc(e), __ATOMIC_RELAXED,
                           __HIP_MEMORY_SCOPE_AGENT);
  }
}

// ---------------------------------------------------------------------------
// K4: unnormalized softmax weights + weighted message scatter-add.
//     alpha = exp(e - max)/sum  ==> accumulate xp[src]*exp(e-max) and sum,
//     normalize per node later (sum is constant per (dst,head)).
// ---------------------------------------------------------------------------
__global__ void k_att_acc(const long long* __restrict__ ei,
                          const float* __restrict__ asrc,
                          const float* __restrict__ adst,
                          const unsigned* __restrict__ emax,
                          const float* __restrict__ xp,
                          float* __restrict__ esum, float* __restrict__ acc,
                          long E_, int N_) {
  long i = (long)blockIdx.x * blockDim.x + threadIdx.x;
  long M = E_ + (long)N_;
  if (i >= M) return;
  int s, d;
  if (i < E_) { s = (int)ei[i]; d = (int)ei[E_ + i]; }
  else        { s = d = (int)(i - E_); }
#pragma unroll
  for (int h = 0; h < 2; ++h) {
    float e = lrelu(asrc[(long)s * 2 + h] + adst[(long)d * 2 + h]);
    float m = fdec(emax[(long)d * 2 + h]);
    float w = __expf(e - m);
    __hip_atomic_fetch_add(&esum[(long)d * 2 + h], w, __ATOMIC_RELAXED,
                           __HIP_MEMORY_SCOPE_AGENT);
    const float* xr = xp + (long)s * 64 + h * 32;
    float* ar = acc + (long)d * 64 + h * 32;
#pragma unroll
    for (int c = 0; c < 32; ++c) {
      __hip_atomic_fetch_add(&ar[c], xr[c] * w, __ATOMIC_RELAXED,
                             __HIP_MEMORY_SCOPE_AGENT);
    }
  }
}

// ---------------------------------------------------------------------------
// K5: h_node = mean over heads of (acc / (esum + 1e-16)) + bias
// ---------------------------------------------------------------------------
__global__ void k_node(const float* __restrict__ acc, const float* __restrict__ esum,
                       const float* __restrict__ bias, float* __restrict__ hn,
                       int N_) {
  long i = (long)blockIdx.x * blockDim.x + threadIdx.x;
  if (i >= (long)N_ * 32) return;
  long n = i >> 5;
  int c = (int)(i & 31);
  float s0 = esum[n * 2 + 0] + 1e-16f;
  float s1 = esum[n * 2 + 1] + 1e-16f;
  float v = 0.5f * (acc[n * 64 + c] / s0 + acc[n * 64 + 32 + c] / s1);
  hn[n * 32 + c] = v + bias[c];
}

// ---------------------------------------------------------------------------
// K6: edge MLP via f32 WMMA, persistent waves.
// B fragments of w1 (64x32) are loaded ONCE into 64 VGPRs and reused across
// all tiles handled by the wave (grid-stride). Per tile: gather
// A[16x64] = [h_node[src] | h_node[dst]] (L2-resident rows), 32 WMMAs,
// hidden -> per-wave LDS slice, then 16 lanes do the 32-MAC second layer.
// Uniform iteration count across all waves => __syncthreads is safe.
// ---------------------------------------------------------------------------
__global__ void k_mlp(const long long* __restrict__ ei, const float* __restrict__ hn,
                      const float* __restrict__ w1, const float* __restrict__ b1,
                      const float* __restrict__ w2, const float* __restrict__ b2,
                      float* __restrict__ out, long E_) {
  __shared__ float hid[8][16][33];  // pad to 33 to avoid LDS bank conflicts
  int wib = threadIdx.x >> 5;       // wave in block (8 waves / 256 threads)
  int lane = threadIdx.x & 31;
  int half = lane >> 4;
  int l16 = lane & 15;

  long numTiles = (E_ + 15) / 16;
  long totalWaves = (long)gridDim.x * (blockDim.x >> 5);
  long waveId = (long)blockIdx.x * (blockDim.x >> 5) + wib;
  long niter = (numTiles + totalWaves - 1) / totalWaves;  // uniform everywhere

  // ---- invariant: B fragments of w1, bias terms (stay in VGPRs) ----
  v2f bf0[16], bf1[16];
#pragma unroll
  for (int t = 0; t < 16; ++t) {
    int k = 4 * t + 2 * half;
    bf0[t].x = w1[(long)k * 32 + l16];
    bf0[t].y = w1[(long)(k + 1) * 32 + l16];
    bf1[t].x = w1[(long)k * 32 + 16 + l16];
    bf1[t].y = w1[(long)(k + 1) * 32 + 16 + l16];
  }
  float bia0 = b1[l16];
  float bia1 = b1[16 + l16];
  float bout = b2[0];

  for (long it = 0; it < niter; ++it) {
    long wt = waveId + it * totalWaves;
    bool active = (wt < numTiles);
    long base = wt * 16;

    long m = active ? (base + l16) : 0;
    if (m >= E_) m = 0;  // clamp; stores guarded below
    int s = (int)ei[m];
    int d = (int)ei[E_ + m];
    const float* rs = hn + (long)s * 32;
    const float* rd = hn + (long)d * 32;

    // prefetch next tile's edge indices into cache
    long nt = wt + totalWaves;
    if (nt < numTiles) {
      __builtin_prefetch(&ei[nt * 16 + l16], 0, 3);
      __builtin_prefetch(&ei[E_ + nt * 16 + l16], 0, 3);
    }

    v8f c0 = {}, c1 = {};
#pragma unroll
    for (int t = 0; t < 16; ++t) {
      int k = 4 * t + 2 * half;  // even => k,k+1 stay within src or dst half
      const float* rp = (k < 32) ? (rs + k) : (rd + (k - 32));
      v2f a;
      a.x = rp[0];
      a.y = rp[1];
      c0 = wmma_f32(a, bf0[t], c0);
      c1 = wmma_f32(a, bf1[t], c1);
    }

#pragma unroll
    for (int r = 0; r < 8; ++r) {
      int mm = r + 8 * half;
      hid[wib][mm][l16]      = fmaxf(c0[r] + bia0, 0.0f);
      hid[wib][mm][16 + l16] = fmaxf(c1[r] + bia1, 0.0f);
    }
    __syncthreads();  // uniform trip count => safe; orders LDS cross-lane

    if (lane < 16) {
      float a2 = bout;
#pragma unroll
      for (int n = 0; n < 32; ++n) a2 += hid[wib][lane][n] * w2[n];
      long e = base + lane;
      if (active && e < E_) out[e] = a2 / TEMPERATURE;
    }
    __syncthreads();  // protect hid before next iteration's writes
  }
}

// ---------------------------------------------------------------------------
extern "C" void kernel_launch(void* const* d_in, const int* in_sizes, int n_in,
                              void* d_out, int out_size, void* d_ws, size_t ws_size,
                              hipStream_t stream) {
  const float* x        = (const float*)d_in[0];
  const long long* ei   = (const long long*)d_in[1];  // int64 [2,E]
  const float* W        = (const float*)d_in[2];
  const float* att_src  = (const float*)d_in[3];
  const float* att_dst  = (const float*)d_in[4];
  const float* bias     = (const float*)d_in[5];
  const float* w1       = (const float*)d_in[6];
  const float* b1       = (const float*)d_in[7];
  const float* w2       = (const float*)d_in[8];
  const float* b2       = (const float*)d_in[9];
  float* out = (float*)d_out;

  const int  N_ = in_sizes[0] / 16;       // IN_C = 16
  const long E_ = (long)in_sizes[1] / 2;  // edge_index is [2,E]

  // workspace layout (floats); total N*168*4B ~= 67 MB
  float* ws    = (float*)d_ws;
  float* xp    = ws;                          // N*64
  float* asrc  = xp + (long)N_ * 64;          // N*2
  float* adst  = asrc + (long)N_ * 2;         // N*2
  unsigned* emax = (unsigned*)(adst + (long)N_ * 2);  // N*2
  float* esum  = (float*)(emax + (long)N_ * 2);       // N*2
  float* acc   = esum + (long)N_ * 2;         // N*64
  float* hn    = acc + (long)N_ * 64;         // N*32

  const int B = 256;
  long M = E_ + (long)N_;

  // K0: init accumulators
  {
    long t = (long)N_ * 64;
    k_init<<<(unsigned)((t + B - 1) / B), B, 0, stream>>>(acc, esum, emax, N_);
  }
  // K1: xp = x @ W (WMMA)
  {
    long waves = ((long)N_ + 15) / 16;
    long t = waves * 32;
    k_xform<<<(unsigned)((t + B - 1) / B), B, 0, stream>>>(x, W, xp, N_);
  }
  // K2: attention coefficients
  {
    long t = (long)N_ * 2;
    k_att_coef<<<(unsigned)((t + B - 1) / B), B, 0, stream>>>(xp, att_src, att_dst,
                                                             asrc, adst, N_);
  }
  // K3: segment max
  k_att_max<<<(unsigned)((M + B - 1) / B), B, 0, stream>>>(ei, asrc, adst, emax, E_, N_);
  // K4: weighted scatter-add
  k_att_acc<<<(unsigned)((M + B - 1) / B), B, 0, stream>>>(ei, asrc, adst, emax, xp,
                                                           esum, acc, E_, N_);
  // K5: node finalize
  {
    long t = (long)N_ * 32;
    k_node<<<(unsigned)((t + B - 1) / B), B, 0, stream>>>(acc, esum, bias, hn, N_);
  }
  // K6: edge MLP (WMMA, persistent waves)
  {
    long numTiles = (E_ + 15) / 16;
    long waves = numTiles;
    long blocks = (waves * 32 + B - 1) / B;
    if (blocks > 1024) blocks = 1024;  // persistent: grid-stride over tiles
    if (blocks < 1) blocks = 1;
    k_mlp<<<(unsigned)blocks, B, 0, stream>>>(ei, hn, w1, b1, w2, b2, out, E_);
  }
}